// IGCN_14714557956429
// MI455X (gfx1250) — compile-verified
//
#include <hip/hip_runtime.h>

typedef __attribute__((ext_vector_type(2))) float v2f;
typedef __attribute__((ext_vector_type(8))) float v8f;

#define N_USERS_C 60000
#define N_NODES_C 100000
#define DFEAT 64

// Native fp32 scatter-add (global_atomic_add_f32, no CAS loop). Results are
// only consumed by later kernels; S_ENDPGM's implicit wait-idle + kernel
// boundary gives the needed visibility.
__device__ __forceinline__ void atomAddF(float* p, float v) {
  unsafeAtomicAdd(p, v);
}

// ---------------- zero fill (float4 grid-stride) ----------------
__global__ void zero_f4(float4* __restrict__ p, long n4) {
  long i = (long)blockIdx.x * blockDim.x + threadIdx.x;
  long stride = (long)gridDim.x * blockDim.x;
  for (; i < n4; i += stride) p[i] = make_float4(0.f, 0.f, 0.f, 0.f);
}

// ---------------- degree count: deg[row[e]] += 1 ----------------
__global__ void degree_count(const int* __restrict__ rows, float* __restrict__ deg, int nE) {
  int e = blockIdx.x * blockDim.x + threadIdx.x;
  if (e < nE) {
    int r = __builtin_nontemporal_load(rows + e);  // streaming index read
    atomAddF(&deg[r], 1.0f);
  }
}

// ---------------- edge-parallel SpMM: xout[row] += w * xin[col] ----------------
// mode 0: w = 1/deg[row]                          (row-normalized feature matrix)
// mode 1: w = rsqrt(deg[row]) * rsqrt(deg[col])   (symmetric adjacency norm)
// 16 lanes per edge, each lane owns a float4 of the 64-wide row.
// Edge index streams are single-use -> non-temporal; feature rows are reused
// heavily (avg degree ~20) -> default temporal so they stay L2-resident.
__global__ void spmm_edge(const int* __restrict__ rows, const int* __restrict__ cols,
                          const float* __restrict__ deg,
                          const float* __restrict__ xin, float* __restrict__ xout,
                          int nE, int mode) {
  int t = blockIdx.x * blockDim.x + threadIdx.x;
  int e = t >> 4;
  int j = t & 15;
  if (e >= nE) return;
  int r = __builtin_nontemporal_load(rows + e);
  int c = __builtin_nontemporal_load(cols + e);
  float w;
  if (mode == 0) {
    w = 1.0f / deg[r];
  } else {
    w = rsqrtf(deg[r]) * rsqrtf(deg[c]);
  }
  const float4* src = (const float4*)(xin + (size_t)c * DFEAT);
  float4 v = src[j];
  float* dst = xout + (size_t)r * DFEAT + (size_t)j * 4;
  atomAddF(dst + 0, w * v.x);
  atomAddF(dst + 1, w * v.y);
  atomAddF(dst + 2, w * v.z);
  atomAddF(dst + 3, w * v.w);
}

// ---------------- acc += rep (float4 grid-stride) ----------------
__global__ void axpy4(float4* __restrict__ acc, const float4* __restrict__ rep, long n4) {
  long i = (long)blockIdx.x * blockDim.x + threadIdx.x;
  long stride = (long)gridDim.x * blockDim.x;
  for (; i < n4; i += stride) {
    float4 a = acc[i];
    float4 b = rep[i];
    a.x += b.x; a.y += b.y; a.z += b.z; a.w += b.w;
    acc[i] = a;
  }
}

// ---------------- gather + epilogue with fp32 WMMA Gram diagonal ----------------
// One wave (32 lanes) per 16-row batch tile. For each of the 3 index sets we
// gather the 16x64 fp32 tile (scaled by 0.25 = mean over L+1 layers), store it
// to d_out, and chain 16x V_WMMA_F32_16X16X4_F32 with A = B = tile registers so
// C accumulates R_u R_u^T + R_p R_p^T + R_n R_n^T; the diagonal of C is l2.
//
// fp32 A 16x4 layout: lane L: VGPR0 = A[L%16][k0 + 2*(L>>4)], VGPR1 = next K.
// fp32 C/D 16x16 layout: VGPR g: lanes 0-15 -> (M=g, N=lane),
//                                 lanes 16-31 -> (M=g+8, N=lane-16).
__global__ void gather_wmma(const float* __restrict__ acc,
                            const int* __restrict__ users,
                            const int* __restrict__ pos,
                            const int* __restrict__ neg,
                            float* __restrict__ out, int Btot) {
  int tile = blockIdx.x;          // Btot/16 tiles
  int lane = threadIdx.x;         // 0..31, one wave per block
  int m  = lane & 15;
  int hi = lane >> 4;             // 0 or 1
  int i  = tile * 16 + m;

  v8f c = {};
  float* l2out = out + (size_t)3 * Btot * DFEAT;

  for (int s = 0; s < 3; ++s) {
    int idx = (s == 0) ? users[i]
            : (s == 1) ? (N_USERS_C + pos[i])
                       : (N_USERS_C + neg[i]);
    const float* rbase = acc + (size_t)idx * DFEAT;
    float* obase = out + (size_t)s * Btot * DFEAT + (size_t)i * DFEAT;
#pragma unroll
    for (int ch = 0; ch < 16; ++ch) {
      int k = ch * 4 + hi * 2;
      v2f a = *(const v2f*)(rbase + k);
      a *= 0.25f;                          // final_rep = acc / (L+1)
      *(v2f*)(obase + k) = a;              // write users_r / pos_r / neg_r
      // D = A * B + C with B registers == A registers -> accumulates R R^T
      c = __builtin_amdgcn_wmma_f32_16x16x4_f32(
          /*neg_a=*/false, a, /*neg_b=*/false, a,
          /*c_mod=*/(short)0, c, /*reuse_a=*/false, /*reuse_b=*/false);
    }
  }

  // extract diagonal: (m,m) for m<8 lives at (vgpr m, lane m);
  // for m>=8 at (vgpr m-8, lane m+16).
  float d = 0.f;
#pragma unroll
  for (int g = 0; g < 8; ++g) {
    if (lane == g || lane == g + 24) d = c[g];
  }
  if (lane < 8)        l2out[tile * 16 + lane]        = d;
  else if (lane >= 24) l2out[tile * 16 + (lane - 16)] = d;
}

// ---------------- host-side orchestration ----------------
extern "C" void kernel_launch(void* const* d_in, const int* in_sizes, int n_in,
                              void* d_out, int out_size, void* d_ws, size_t ws_size,
                              hipStream_t stream) {
  const float* emb     = (const float*)d_in[0];
  const int* feat_row  = (const int*)d_in[1];
  const int* feat_col  = (const int*)d_in[2];
  const int* adj_row   = (const int*)d_in[3];
  const int* adj_col   = (const int*)d_in[4];
  const int* users     = (const int*)d_in[5];
  const int* pos_items = (const int*)d_in[6];
  const int* neg_items = (const int*)d_in[7];

  const int F = in_sizes[1];   // 2,100,000 feat edges
  const int A = in_sizes[3];   // 2,000,000 adj edges
  const int B = in_sizes[5];   // 4096

  const size_t degSlot = 100352;                      // N_NODES rounded up to 2048
  const size_t matSlot = (size_t)N_NODES_C * DFEAT;   // 6,400,000 floats

  float* ws    = (float*)d_ws;
  float* f_deg = ws;
  float* a_deg = f_deg + degSlot;
  float* ACC   = a_deg + degSlot;
  float* RA    = ACC + matSlot;
  float* RB    = RA + matSlot;
  const long totalF = (long)(2 * degSlot + 3 * matSlot);

  const int ZB = 2048, ZT = 256;

  // 1) zero degrees + ACC + RA + RB
  zero_f4<<<ZB, ZT, 0, stream>>>((float4*)ws, totalF / 4);

  // 2) degrees
  degree_count<<<(F + 255) / 256, 256, 0, stream>>>(feat_row, f_deg, F);
  degree_count<<<(A + 255) / 256, 256, 0, stream>>>(adj_row, a_deg, A);

  // 3) x = D_row^{-1} F emb  -> ACC (acc starts as x)
  spmm_edge<<<(F * 16 + 255) / 256, 256, 0, stream>>>(feat_row, feat_col, f_deg,
                                                      emb, ACC, F, 0);

  const int AG = (A * 16 + 255) / 256;
  const long n4 = (long)matSlot / 4;

  // layer 1: RA = Adj*ACC ; ACC += RA
  spmm_edge<<<AG, 256, 0, stream>>>(adj_row, adj_col, a_deg, ACC, RA, A, 1);
  axpy4<<<ZB, ZT, 0, stream>>>((float4*)ACC, (const float4*)RA, n4);

  // layer 2: RB = Adj*RA ; ACC += RB
  spmm_edge<<<AG, 256, 0, stream>>>(adj_row, adj_col, a_deg, RA, RB, A, 1);
  axpy4<<<ZB, ZT, 0, stream>>>((float4*)ACC, (const float4*)RB, n4);

  // layer 3: RA = 0 ; RA = Adj*RB ; ACC += RA
  zero_f4<<<ZB, ZT, 0, stream>>>((float4*)RA, n4);
  spmm_edge<<<AG, 256, 0, stream>>>(adj_row, adj_col, a_deg, RB, RA, A, 1);
  axpy4<<<ZB, ZT, 0, stream>>>((float4*)ACC, (const float4*)RA, n4);

  // 4) gather batch rows, scale by 1/4, emit l2 via fp32 WMMA Gram diagonal
  gather_wmma<<<B / 16, 32, 0, stream>>>(ACC, users, pos_items, neg_items,
                                         (float*)d_out, B);
}